// embed_graph_36060545417378
// MI455X (gfx1250) — compile-verified
//
#include <hip/hip_runtime.h>

#define B    4
#define D    3
#define N    48
#define NPAIR 1128
#define C    2256          // ordered pairs
#define CP   2272          // C padded to 71*32 (branch-free A-tile loads)
#define NE   46            // complement size
#define E1   277           // d_out_comb
#define DC   286           // d_comb
#define DCP  288           // padded K (multiple of 4)
#define E2   289           // d_graph
#define E2P  320           // E2 padded to 10*32 (branch-free B-tile loads)
#define NT2  71            // CP/32  n-tile pairs
#define ET2  10            // E2P/32 e-tile pairs
#define SORT_N 4096

typedef float v2f __attribute__((ext_vector_type(2)));
typedef float v8f __attribute__((ext_vector_type(8)));

// ---------------- kernel 1: center x over point axis ----------------
__global__ void k_center(const float* __restrict__ x, float* __restrict__ X) {
    int row = threadIdx.x;                      // 0..B*D-1
    if (row < B * D) {
        const float* src = x + row * N;
        float s = 0.f;
        for (int t = 0; t < N; ++t) s += src[t];
        float m = s * (1.0f / N);
        float* dst = X + row * N;
        for (int t = 0; t < N; ++t) dst[t] = src[t] - m;
    }
}

// ---- kernel 2: build padded A_graph, zero G pad rows/columns -------
__global__ void k_prep(const float* __restrict__ Ag, float* __restrict__ AgP,
                       float* __restrict__ G) {
    int idx = blockIdx.x * blockDim.x + threadIdx.x;
    // padded A_graph (DCP x E2P), zeros outside (DC x E2)
    if (idx < DCP * E2P) {
        int d = idx / E2P, e = idx % E2P;
        AgP[idx] = (d < DC && e < E2) ? Ag[d * E2 + e] : 0.0f;
    }
    // zero G pad columns n = C..CP-1 for all rows
    if (idx < B * DCP * (CP - C)) {
        int b = idx / (DCP * (CP - C));
        int r = idx % (DCP * (CP - C));
        int d = r / (CP - C);
        int n = C + r % (CP - C);
        G[((size_t)b * DCP + d) * CP + n] = 0.0f;
    }
    // zero G pad rows d = DC..DCP-1 for all columns
    if (idx < B * (DCP - DC) * CP) {
        int b = idx / ((DCP - DC) * CP);
        int r = idx % ((DCP - DC) * CP);
        int d = DC + r / CP;
        int n = r % CP;
        G[((size_t)b * DCP + d) * CP + n] = 0.0f;
    }
}

// ---- kernel 3: per (b, pair): M, MtM -> vec1, MtV, embedding 1 -----
__global__ void __launch_bounds__(288) k_pairs(
        const float* __restrict__ X, const float* __restrict__ A_comb,
        const float* __restrict__ w_comb, float* __restrict__ G) {
    __shared__ float Xs[D][N];
    __shared__ float Ms[3][3];      // Ms[j][k] : column k = {X_i, X_j, X_i x X_j}
    __shared__ float MtV[3][NE];

    int bc = blockIdx.x;
    int b  = bc / C;
    int c  = bc % C;
    int t  = threadIdx.x;

    if (t < D * N) Xs[t / N][t % N] = X[b * D * N + t];
    __syncthreads();

    // decode ordered pair (i, j)
    int cc = (c < NPAIR) ? c : c - NPAIR;
    int pi = 0, rem = cc;
    while (rem >= (N - 1 - pi)) { rem -= (N - 1 - pi); ++pi; }
    int pj = pi + 1 + rem;
    if (c >= NPAIR) { int tmp = pi; pi = pj; pj = tmp; }
    int lo = (pi < pj) ? pi : pj;
    int hi = (pi < pj) ? pj : pi;

    if (t < 9) {
        int jj = t / 3, k = t % 3;
        float val;
        if (k == 0)      val = Xs[jj][pi];
        else if (k == 1) val = Xs[jj][pj];
        else {
            int a1 = (jj + 1) % 3, a2 = (jj + 2) % 3;
            val = Xs[a1][pi] * Xs[a2][pj] - Xs[a2][pi] * Xs[a1][pj];
        }
        Ms[jj][k] = val;
    }
    __syncthreads();

    // vec1 = MtM (row-major k*3+l) -> G rows 0..8
    if (t < 9) {
        int k = t / 3, l = t % 3;
        float s = Ms[0][k] * Ms[0][l] + Ms[1][k] * Ms[1][l] + Ms[2][k] * Ms[2][l];
        G[((size_t)b * DCP + t) * CP + c] = s;
    }
    // MtV[k][m] over complement indices
    if (t < 3 * NE) {
        int k = t / NE, m = t % NE;
        int idx = m;
        if (idx >= lo) ++idx;
        if (idx >= hi) ++idx;
        MtV[k][m] = Ms[0][k] * Xs[0][idx] + Ms[1][k] * Xs[1][idx] + Ms[2][k] * Xs[2][idx];
    }
    __syncthreads();

    // embedding 1: one thread per output feature e
    if (t < E1) {
        float a0 = A_comb[0 * E1 + t];
        float a1 = A_comb[1 * E1 + t];
        float a2 = A_comb[2 * E1 + t];
        float p[NE];
#pragma unroll
        for (int m = 0; m < NE; ++m)
            p[m] = MtV[0][m] * a0 + MtV[1][m] * a1 + MtV[2][m] * a2;

        // Batcher odd-even mergesort network for 64 slots; comparators that
        // touch index >= NE are pruned (virtual +inf elements never move).
#pragma unroll
        for (int pp = 1; pp < 64; pp <<= 1) {
#pragma unroll
            for (int kk = pp; kk >= 1; kk >>= 1) {
#pragma unroll
                for (int jj = kk & (pp - 1); jj + kk < 64; jj += 2 * kk) {
#pragma unroll
                    for (int ii = 0; ii < kk; ++ii) {
                        int a = ii + jj, bq = ii + jj + kk;
                        if (bq < NE && (a / (2 * pp)) == (bq / (2 * pp))) {
                            float vlo = fminf(p[a], p[bq]);
                            float vhi = fmaxf(p[a], p[bq]);
                            p[a] = vlo; p[bq] = vhi;
                        }
                    }
                }
            }
        }

        float acc = 0.f;
#pragma unroll
        for (int m = 0; m < NE; ++m)
            acc += p[m] * w_comb[m * E1 + t];
        G[((size_t)b * DCP + 9 + t) * CP + c] = acc;
    }
}

// ----- kernel 4: WMMA f32 GEMM  P[b,e,n] = sum_d G[b,d,n]*Ag[d,e] -----
// One wave per 32x32 output block (2x2 register-blocked 16x16 WMMA tiles).
// Operand buffers are zero-padded so the K-loop is branch-free:
// 8 loads + 4 WMMAs per step, A/B fragments each reused twice.
__global__ void __launch_bounds__(32) k_wmma(
        const float* __restrict__ G, const float* __restrict__ AgP,
        float* __restrict__ P) {
    int tile = blockIdx.x;
    int b    = blockIdx.y;
    int n0   = (tile % NT2) * 32;
    int e0   = (tile / NT2) * 32;
    int lane = threadIdx.x;
    int half = lane >> 4;         // lanes 16-31 carry K+2 / M+8
    int l    = lane & 15;

    const float* Gb = G + (size_t)b * DCP * CP;
    v8f acc00 = {}, acc01 = {}, acc10 = {}, acc11 = {};

    for (int d0 = 0; d0 < DCP; d0 += 4) {
        int dA = d0 + 2 * half;
        v2f a0, a1, b0, b1;
        a0.x = Gb[(size_t)(dA + 0) * CP + n0 + l];
        a0.y = Gb[(size_t)(dA + 1) * CP + n0 + l];
        a1.x = Gb[(size_t)(dA + 0) * CP + n0 + 16 + l];
        a1.y = Gb[(size_t)(dA + 1) * CP + n0 + 16 + l];
        b0.x = AgP[(dA + 0) * E2P + e0 + l];
        b0.y = AgP[(dA + 1) * E2P + e0 + l];
        b1.x = AgP[(dA + 0) * E2P + e0 + 16 + l];
        b1.y = AgP[(dA + 1) * E2P + e0 + 16 + l];

        acc00 = __builtin_amdgcn_wmma_f32_16x16x4_f32(false, a0, false, b0, (short)0, acc00, false, false);
        acc01 = __builtin_amdgcn_wmma_f32_16x16x4_f32(false, a0, false, b1, (short)0, acc01, false, false);
        acc10 = __builtin_amdgcn_wmma_f32_16x16x4_f32(false, a1, false, b0, (short)0, acc10, false, false);
        acc11 = __builtin_amdgcn_wmma_f32_16x16x4_f32(false, a1, false, b1, (short)0, acc11, false, false);
    }

    // C/D layout: VGPR r -> row n+r (+8 for upper lanes), col e+l
    v8f accs[4] = {acc00, acc01, acc10, acc11};
#pragma unroll
    for (int nb = 0; nb < 2; ++nb) {
#pragma unroll
        for (int eb = 0; eb < 2; ++eb) {
            int nbase = n0 + nb * 16;
            int e = e0 + eb * 16 + l;
            if (nbase < C && e < E2) {
                v8f acc = accs[nb * 2 + eb];
                float* dst = P + ((size_t)b * E2 + e) * C + nbase + 8 * half;
#pragma unroll
                for (int r = 0; r < 8; ++r) dst[r] = acc[r];
            }
        }
    }
}

// --- kernel 5: per (b,e): bitonic sort 2256 vals in LDS + weighted sum ---
__global__ void __launch_bounds__(512) k_sort_reduce(
        const float* __restrict__ P, const float* __restrict__ w_graph,
        float* __restrict__ out) {
    __shared__ float s[SORT_N];
    __shared__ float red[512];
    int be = blockIdx.x;
    int b  = be / E2, e = be % E2;
    int t  = threadIdx.x;

    const float* src = P + ((size_t)b * E2 + e) * C;
    for (int idx = t; idx < SORT_N; idx += 512)
        s[idx] = (idx < C) ? src[idx] : __builtin_inff();
    __syncthreads();

    for (int k = 2; k <= SORT_N; k <<= 1) {
        for (int jj = k >> 1; jj > 0; jj >>= 1) {
            for (int idx = t; idx < SORT_N; idx += 512) {
                int ixj = idx ^ jj;
                if (ixj > idx) {
                    float A0 = s[idx], B0 = s[ixj];
                    bool up = ((idx & k) == 0);
                    if ((A0 > B0) == up) { s[idx] = B0; s[ixj] = A0; }
                }
            }
            __syncthreads();
        }
    }

    float acc = 0.f;
    for (int cIdx = t; cIdx < C; cIdx += 512)
        acc += s[cIdx] * w_graph[cIdx * E2 + e];
    red[t] = acc;
    __syncthreads();
    for (int off = 256; off > 0; off >>= 1) {
        if (t < off) red[t] += red[t + off];
        __syncthreads();
    }
    if (t == 0) out[b * E2 + e] = red[0];
}

extern "C" void kernel_launch(void* const* d_in, const int* in_sizes, int n_in,
                              void* d_out, int out_size, void* d_ws, size_t ws_size,
                              hipStream_t stream) {
    const float* x       = (const float*)d_in[0];
    const float* A_comb  = (const float*)d_in[1];
    const float* w_comb  = (const float*)d_in[2];
    const float* A_graph = (const float*)d_in[3];
    const float* w_graph = (const float*)d_in[4];
    float* out = (float*)d_out;

    // workspace layout (floats): X | G (B x DCP x CP) | AgP (DCP x E2P) | P (B x E2 x C)
    float* X   = (float*)d_ws;
    float* G   = X + B * D * N;
    float* AgP = G + (size_t)B * DCP * CP;
    float* P   = AgP + (size_t)DCP * E2P;

    k_center<<<1, 32, 0, stream>>>(x, X);
    k_prep<<<(DCP * E2P + 255) / 256, 256, 0, stream>>>(A_graph, AgP, G);
    k_pairs<<<B * C, 288, 0, stream>>>(X, A_comb, w_comb, G);
    dim3 gD(NT2 * ET2, B);
    k_wmma<<<gD, 32, 0, stream>>>(G, AgP, P);
    k_sort_reduce<<<B * E2, 512, 0, stream>>>(P, w_graph, out);
}